// FixedVAR_48601849921953
// MI455X (gfx1250) — compile-verified
//
#include <hip/hip_runtime.h>
#include <stdint.h>

typedef float v2f __attribute__((ext_vector_type(2)));
typedef float v8f __attribute__((ext_vector_type(8)));

#define WAVES    8
#define TSTRIDE  68   // floats: rows 272B (16B aligned), lane skew 4 banks -> conflict-free b64 reads
#define PSTRIDE  18   // floats: even -> 8B-aligned v2f reloads of P

// One wave copies a 16x64 f32 tile (4KB) global -> LDS, fully coalesced:
// 8 async ops x (32 lanes x 16B) = 8 x 512B. Tracked by ASYNCcnt.
__device__ __forceinline__ void tile_async_16x64(uint32_t lds_base_bytes,
                                                 const float* gbase,
                                                 int hi, int ln4) {
    #pragma unroll
    for (int t = 0; t < 8; ++t) {
        const int row = 2 * t + hi;                       // lanes 0-15 row 2t, 16-31 row 2t+1
        const uint32_t loff = lds_base_bytes
                            + (uint32_t)(row * TSTRIDE) * 4u
                            + (uint32_t)ln4 * 4u;
        const float* g = gbase + (size_t)row * 1024 + ln4;
        asm volatile("global_load_async_to_lds_b128 %0, %1, off"
                     :: "v"(loff), "v"(g)
                     : "memory");
    }
}

__device__ __forceinline__ void wait_async_le8() {
    asm volatile("s_wait_asynccnt 0x8" ::: "memory");
}
__device__ __forceinline__ void wait_async_0() {
    asm volatile("s_wait_asynccnt 0x0" ::: "memory");
}

__global__ __launch_bounds__(256) void fixedvar_attn_f32wmma(
    const float* __restrict__ X,   // (B, L, D) = (32, 128, 1024), K source
    const float* __restrict__ FQ,  // (1, C, L, H, d) = (8, 128, 16, 64)
    const float* __restrict__ FV,  // same shape as FQ
    float* __restrict__ Y)         // (B, L, D)
{
    constexpr int L = 128, D = 1024, dd = 64, C = 8;

    __shared__ __align__(16) float sQ[WAVES][16 * TSTRIDE];
    __shared__ __align__(16) float sK[WAVES][2][16 * TSTRIDE]; // double-buffered K
    __shared__ __align__(16) float sP[WAVES][16 * PSTRIDE];

    const int lane = threadIdx.x & 31;
    const int warp = threadIdx.x >> 5;

    // one wave per (b, h, i): 512 blocks x 8 waves = 4096 waves, no redundant math,
    // each block holds i = 0..7 of one (b,h) -> 36 inner tiles per block (balanced)
    const int bh = blockIdx.x;          // 0..511
    const int i  = warp;                // Q row-tile index, 0..7
    const int h  = bh & 15;
    const int b  = bh >> 4;
    const int c  = b & (C - 1);

    const int ln  = lane & 15;          // A: M-row / B: N-col / C,D: col
    const int hi  = lane >> 4;          // half-wave select
    const int ko  = hi * 2;             // K offset in f32 A/B operand layout
    const int ln4 = ln * 4;

    const float* Kp = X  + (size_t)b * L * D + (size_t)h * dd;
    const float* Qp = FQ + (size_t)c * L * D + (size_t)h * dd; // H*d == D
    const float* Vp = FV + (size_t)c * L * D + (size_t)h * dd;
    float*       Op = Y  + (size_t)b * L * D + (size_t)h * dd;

    float* qt = &sQ[warp][0];
    float* pt = &sP[warp][0];
    float* kt[2] = { &sK[warp][0][0], &sK[warp][1][0] };
    const uint32_t qoff  = (uint32_t)(uintptr_t)qt;       // low 32b of generic ptr == LDS byte addr
    const uint32_t koff0 = (uint32_t)(uintptr_t)kt[0];
    const uint32_t koff1 = (uint32_t)(uintptr_t)kt[1];

    // stage Q_i and K_0 back-to-back (16 async ops in flight)
    tile_async_16x64(qoff,  Qp + (size_t)(i * 16) * D, hi, ln4);
    tile_async_16x64(koff0, Kp, hi, ln4);

    // Q done when <=8 outstanding (async loads complete in order)
    wait_async_le8();
    v2f q[16];
    #pragma unroll
    for (int s = 0; s < 16; ++s)
        q[s] = *(const v2f*)(qt + ln * TSTRIDE + 4 * s + ko);

    v8f y0 = {0.f,0.f,0.f,0.f,0.f,0.f,0.f,0.f};
    v8f y1 = y0, y2 = y0, y3 = y0;

    for (int j = 0; j <= i; ++j) {
        // prefetch K_{j+1} into the other buffer, then wait for K_j
        if (j < i) {
            tile_async_16x64((j & 1) ? koff0 : koff1,
                             Kp + (size_t)((j + 1) * 16) * D, hi, ln4);
            wait_async_le8();   // K_j complete, K_{j+1} still in flight
        } else {
            wait_async_0();     // last tile: drain everything
        }

        const float* ktile = kt[j & 1];

        // K_j rows read with the A-operand pattern == K^T in B-operand layout
        v2f kb[16];
        #pragma unroll
        for (int s = 0; s < 16; ++s)
            kb[s] = *(const v2f*)(ktile + ln * TSTRIDE + 4 * s + ko);

        // P = Q_i @ K_j^T : 16x16 f32 accum, K=64 via 16 wmma steps
        v8f p = {0.f,0.f,0.f,0.f,0.f,0.f,0.f,0.f};
        #pragma unroll
        for (int s = 0; s < 16; ++s)
            p = __builtin_amdgcn_wmma_f32_16x16x4_f32(
                    false, q[s], false, kb[s], (short)0, p, false, false);

        // causal mask on the diagonal tile; park P in LDS (same-wave DS ops in order)
        const bool diag = (j == i);
        #pragma unroll
        for (int v = 0; v < 8; ++v) {
            const int r = v + 8 * hi;           // C/D row
            float val = p[v];
            if (diag && (ln > r)) val = 0.f;    // col > row -> masked
            pt[r * PSTRIDE + ln] = val;
        }

        // reload P in A-operand layout
        v2f pa[4];
        #pragma unroll
        for (int s = 0; s < 4; ++s)
            pa[s] = *(const v2f*)(pt + ln * PSTRIDE + 4 * s + ko);

        // Y_i += P @ V_j : N=64 as 4 tiles, K=16 as 4 steps (V loads are 2x64B coalesced)
        const float* vbase = Vp + (size_t)(j * 16) * D;
        #pragma unroll
        for (int s = 0; s < 4; ++s) {
            const float* vr0 = vbase + (size_t)(4 * s + ko) * D + ln;
            const float* vr1 = vr0 + D;
            v2f vb;
            vb.x = vr0[0];   vb.y = vr1[0];
            y0 = __builtin_amdgcn_wmma_f32_16x16x4_f32(
                     false, pa[s], false, vb, (short)0, y0, false, false);
            vb.x = vr0[16];  vb.y = vr1[16];
            y1 = __builtin_amdgcn_wmma_f32_16x16x4_f32(
                     false, pa[s], false, vb, (short)0, y1, false, false);
            vb.x = vr0[32];  vb.y = vr1[32];
            y2 = __builtin_amdgcn_wmma_f32_16x16x4_f32(
                     false, pa[s], false, vb, (short)0, y2, false, false);
            vb.x = vr0[48];  vb.y = vr1[48];
            y3 = __builtin_amdgcn_wmma_f32_16x16x4_f32(
                     false, pa[s], false, vb, (short)0, y3, false, false);
        }
    }

    // store Y_i from C/D layout (rows 64B-contiguous per half-wave)
    #pragma unroll
    for (int v = 0; v < 8; ++v) {
        const int r = i * 16 + v + 8 * hi;
        float* orow = Op + (size_t)r * D + ln;
        orow[0]  = y0[v];
        orow[16] = y1[v];
        orow[32] = y2[v];
        orow[48] = y3[v];
    }
}

extern "C" void kernel_launch(void* const* d_in, const int* in_sizes, int n_in,
                              void* d_out, int out_size, void* d_ws, size_t ws_size,
                              hipStream_t stream) {
    const float* X  = (const float*)d_in[0];
    const float* FQ = (const float*)d_in[1];
    const float* FV = (const float*)d_in[2];
    float* Y = (float*)d_out;
    // 512 blocks x 8 wave32s: one wave per (batch, head, row-tile)
    fixedvar_attn_f32wmma<<<512, 256, 0, stream>>>(X, FQ, FV, Y);
}